// HIPABlock_42752104465010
// MI455X (gfx1250) — compile-verified
//
#include <hip/hip_runtime.h>
#include <hip/hip_bf16.h>
#include <math.h>

typedef __attribute__((ext_vector_type(2))) float v2f;
typedef __attribute__((ext_vector_type(8))) float v8f;

#define B_    4
#define C_    256
#define H_    256
#define W_    256
#define COUT_ 256

// workspace layout (float offsets)
#define PYR3_OFF 0         // [B][C][64]
#define PYR2_OFF 65536     // [B][C][16]
#define PYR1_OFF 81920     // [B][C][4]
#define PYR0_OFF 86016     // [B][C][1]
#define WS_FLOATS 87040
// int region: idxbuf [B][4][32] (512), then tbl [B][96] (384)

// d_out layout (float offsets)
#define OUT_OFF      0
#define SPARSE_OFF   67108864   // [128][256]
#define COORD_OFF    67141632   // [128][4]
#define SPARSITY_OFF 67142144

// ---------------------------------------------------------------------------
// Kernel 1: one block per (b,c) slice. Fully coalesced read of x (each wave
// reads contiguous 1KB rows), produces 8x8 cell maxes, folds to 4x4/2x2/1x1
// in LDS.
// ---------------------------------------------------------------------------
__global__ void k_pyramid(const float* __restrict__ x, float* __restrict__ wsf) {
  int bc = blockIdx.x;                       // b*C + c
  const float* xp = x + (size_t)bc * (H_ * W_);
  int t = threadIdx.x;
  int cell = t >> 2, sub = t & 3;            // 4 threads per 32x32 cell
  int cy = cell >> 3, cx = cell & 7;
  const float* base = xp + (cy * 32) * W_ + cx * 32 + sub * 8;
  float m = -INFINITY;
  for (int r = 0; r < 32; ++r) {
    const float4* p = (const float4*)(base + r * W_);
    float4 a = p[0], b = p[1];
    m = fmaxf(m, fmaxf(fmaxf(a.x, a.y), fmaxf(a.z, a.w)));
    m = fmaxf(m, fmaxf(fmaxf(b.x, b.y), fmaxf(b.z, b.w)));
  }
  __shared__ float red[256];
  __shared__ float s3[64], s2[16], s1[4];
  red[t] = m;
  __syncthreads();
  if (sub == 0) {
    float cm = fmaxf(fmaxf(red[t], red[t + 1]), fmaxf(red[t + 2], red[t + 3]));
    s3[cell] = cm;
    wsf[PYR3_OFF + bc * 64 + cell] = cm;
  }
  __syncthreads();
  if (t < 16) {
    int y = t >> 2, xk = t & 3;
    float cm = fmaxf(fmaxf(s3[(2 * y) * 8 + 2 * xk], s3[(2 * y) * 8 + 2 * xk + 1]),
                     fmaxf(s3[(2 * y + 1) * 8 + 2 * xk], s3[(2 * y + 1) * 8 + 2 * xk + 1]));
    s2[t] = cm;
    wsf[PYR2_OFF + bc * 16 + t] = cm;
  }
  __syncthreads();
  if (t < 4) {
    int y = t >> 1, xk = t & 1;
    float cm = fmaxf(fmaxf(s2[(2 * y) * 4 + 2 * xk], s2[(2 * y) * 4 + 2 * xk + 1]),
                     fmaxf(s2[(2 * y + 1) * 4 + 2 * xk], s2[(2 * y + 1) * 4 + 2 * xk + 1]));
    s1[t] = cm;
    wsf[PYR1_OFF + bc * 4 + t] = cm;
  }
  __syncthreads();
  if (t == 0) {
    wsf[PYR0_OFF + bc] = fmaxf(fmaxf(s1[0], s1[1]), fmaxf(s1[2], s1[3]));
  }
}

// ---------------------------------------------------------------------------
// Kernel 2: one block per (b, level). l2 norms + importance + serial top-k
// (N<=64, k<=19; descending, ties -> lowest index, matching lax.top_k).
// Also writes the paint lookup tables, coords, and sparsity scalar.
// ---------------------------------------------------------------------------
__global__ void k_topk(const float* __restrict__ wsf, int* __restrict__ idxbuf,
                       int* __restrict__ tbl, float* __restrict__ out) {
  int b = blockIdx.x >> 2, level = blockIdx.x & 3;
  int g = 1 << level, N = g * g;
  const int kk[4]     = {1, 4, 8, 19};
  const int off[4]    = {0, 1, 5, 13};
  const int pyroff[4] = {PYR0_OFF, PYR1_OFF, PYR2_OFF, PYR3_OFF};
  const int tbloff[4] = {84, 80, 64, 0};
  int k = kk[level];
  __shared__ float impbuf[64];
  int t = threadIdx.x;
  int* tblb = tbl + b * 96;
  if (t < N) {
    const float* pl = wsf + pyroff[level];
    float ss = 0.f;
    for (int c = 0; c < C_; ++c) { float v = pl[(b * C_ + c) * N + t]; ss += v * v; }
    float l2 = sqrtf(ss);
    float imp;
    if (level < 3) {  // |l2 - parent_l2| using stride-2 subsample of finer level
      const float* pp = wsf + pyroff[level + 1];
      int y = t / g, xk = t % g;
      int p = (2 * y) * (2 * g) + 2 * xk;
      int Np = N * 4;
      float sp = 0.f;
      for (int c = 0; c < C_; ++c) { float v = pp[(b * C_ + c) * Np + p]; sp += v * v; }
      imp = fabsf(l2 - sqrtf(sp));
    } else {
      imp = l2;
    }
    impbuf[t] = imp;
    tblb[tbloff[level] + t] = -1;
  }
  __syncthreads();
  if (t == 0) {
    for (int r = 0; r < k; ++r) {
      float bv = -INFINITY; int bi = 0;
      for (int n = 0; n < N; ++n)
        if (impbuf[n] > bv) { bv = impbuf[n]; bi = n; }
      impbuf[bi] = -INFINITY;
      int row = b * 32 + off[level] + r;
      idxbuf[(b * 4 + level) * 32 + r] = bi;
      tblb[tbloff[level] + bi] = row;
      int iy = bi / g, ix = bi % g;
      float* co = out + COORD_OFF + row * 4;
      co[0] = (ix + 0.5f) / g; co[1] = (iy + 0.5f) / g;
      co[2] = 1.0f / g;        co[3] = 1.0f / g;
    }
    if (b == 0 && level == 0) out[SPARSITY_OFF] = 32.0f / 65536.0f;
  }
}

// ---------------------------------------------------------------------------
// Kernel 3: gather + LayerNorm + GEMM on the fp32 WMMA pipe.
// 8 blocks (one 16-row M-tile each), 8 waves/block, 2 N-tiles/wave,
// K=256 in steps of 4 -> 128 v_wmma_f32_16x16x4_f32 per wave.
// ---------------------------------------------------------------------------
__global__ void k_proj(const float* __restrict__ wsf, const int* __restrict__ idxbuf,
                       const float* __restrict__ lng, const float* __restrict__ lnb,
                       const float* __restrict__ wmat, const float* __restrict__ bias,
                       float* __restrict__ out) {
  __shared__ float As[16][257];   // stride 257 breaks bank conflicts on A-frag reads
  const int off[4]    = {0, 1, 5, 13};
  const int pyroff[4] = {PYR0_OFF, PYR1_OFF, PYR2_OFF, PYR3_OFF};
  int t = threadIdx.x;
  int wv = t >> 5, l = t & 31;
  int mtile = blockIdx.x;

  // Phase A: each wave LayerNorms 2 tokens into LDS
  for (int s = 0; s < 2; ++s) {
    int tok = wv * 2 + s;
    int row = mtile * 16 + tok;
    int b = row >> 5, j = row & 31;
    int level = (j < 1) ? 0 : (j < 5) ? 1 : (j < 13) ? 2 : 3;
    int rank = j - off[level];
    int cell = idxbuf[(b * 4 + level) * 32 + rank];
    int N = 1 << (2 * level);
    const float* pl = wsf + pyroff[level];
    float vv[8];
    float sum1 = 0.f, sumsq = 0.f;
    for (int i = 0; i < 8; ++i) {
      int c = l + 32 * i;
      float v = pl[(b * C_ + c) * N + cell];
      vv[i] = v; sum1 += v; sumsq += v * v;
    }
    for (int m = 16; m >= 1; m >>= 1) {
      sum1  += __shfl_xor(sum1, m, 32);
      sumsq += __shfl_xor(sumsq, m, 32);
    }
    float mean = sum1 * (1.0f / 256.0f);
    float var  = sumsq * (1.0f / 256.0f) - mean * mean;
    float rstd = rsqrtf(var + 1e-5f);
    for (int i = 0; i < 8; ++i) {
      int c = l + 32 * i;
      As[tok][c] = (vv[i] - mean) * rstd * lng[c] + lnb[c];
    }
  }
  __syncthreads();

  // Phase B: WMMA GEMM. ISA layout: A lanes0-15 hold K0/K1 of M=lane,
  // lanes16-31 hold K2/K3; B/C/D row-striped across lanes.
  int nb0 = wv * 32, nb1 = wv * 32 + 16;
  v8f acc0 = {}, acc1 = {};
  int mrow = l & 15;
  int koff = (l >> 4) << 1;
  int ncol = l & 15;
  for (int k0 = 0; k0 < 256; k0 += 4) {
    v2f a;
    a.x = As[mrow][k0 + koff];
    a.y = As[mrow][k0 + koff + 1];
    const float* wr0 = wmat + (size_t)(k0 + koff) * COUT_;
    const float* wr1 = wmat + (size_t)(k0 + koff + 1) * COUT_;
    v2f b0, b1;
    b0.x = wr0[nb0 + ncol]; b0.y = wr1[nb0 + ncol];
    b1.x = wr0[nb1 + ncol]; b1.y = wr1[nb1 + ncol];
    acc0 = __builtin_amdgcn_wmma_f32_16x16x4_f32(false, a, false, b0, (short)0, acc0, false, false);
    acc1 = __builtin_amdgcn_wmma_f32_16x16x4_f32(false, a, false, b1, (short)0, acc1, false, false);
  }

  // Phase C: add bias, write sparse_seq rows (also consumed by paint kernel)
  int mhi = (l >> 4) * 8;
  for (int vr = 0; vr < 8; ++vr) {
    int row = mtile * 16 + vr + mhi;
    int n0 = nb0 + ncol, n1 = nb1 + ncol;
    out[SPARSE_OFF + row * COUT_ + n0] = acc0[vr] + bias[n0];
    out[SPARSE_OFF + row * COUT_ + n1] = acc1[vr] + bias[n1];
  }
}

// ---------------------------------------------------------------------------
// Kernel 4: dense paint. Per float4 of out, resolve finest selected covering
// cell (level-0 always covers -> every pixel written). Coalesced 256 MiB store.
// ---------------------------------------------------------------------------
__global__ void k_paint(const int* __restrict__ tbl, const float* __restrict__ sparse,
                        float* __restrict__ out) {
  size_t q = (size_t)blockIdx.x * blockDim.x + threadIdx.x;
  size_t e = q * 4;                         // out element index, x-aligned to 4
  int x  = (int)(e & 255);
  int y  = (int)((e >> 8) & 255);
  int co = (int)((e >> 16) & 255);
  int b  = (int)(e >> 24);
  const int* tblb = tbl + b * 96;
  int r = tblb[((y >> 5) << 3) + (x >> 5)];           // level 3
  if (r < 0) {
    r = tblb[64 + ((y >> 6) << 2) + (x >> 6)];        // level 2
    if (r < 0) {
      r = tblb[80 + ((y >> 7) << 1) + (x >> 7)];      // level 1
      if (r < 0) r = tblb[84];                        // level 0 (always selected)
    }
  }
  float val = sparse[r * COUT_ + co];
  ((float4*)out)[q] = make_float4(val, val, val, val);
}

extern "C" void kernel_launch(void* const* d_in, const int* in_sizes, int n_in,
                              void* d_out, int out_size, void* d_ws, size_t ws_size,
                              hipStream_t stream) {
  const float* x   = (const float*)d_in[0];
  const float* lng = (const float*)d_in[1];
  const float* lnb = (const float*)d_in[2];
  const float* wm  = (const float*)d_in[3];
  const float* bs  = (const float*)d_in[4];
  float* out = (float*)d_out;
  float* wsf = (float*)d_ws;
  int* wsi = (int*)(wsf + WS_FLOATS);
  int* idxbuf = wsi;          // [B][4][32]
  int* tbl    = wsi + 512;    // [B][96]

  k_pyramid<<<B_ * C_, 256, 0, stream>>>(x, wsf);
  k_topk<<<B_ * 4, 256, 0, stream>>>(wsf, idxbuf, tbl, out);
  k_proj<<<8, 256, 0, stream>>>(wsf, idxbuf, lng, lnb, wm, bs, out);
  k_paint<<<65536, 256, 0, stream>>>(tbl, out + SPARSE_OFF, out);
}